// InterDistanceLossAdjacent_5179730559712
// MI455X (gfx1250) — compile-verified
//
#include <hip/hip_runtime.h>
#include <hip/hip_bf16.h>

// InterDistanceLossAdjacent for MI455X (gfx1250, wave32).
// dist^2(i,j) = |xi|^2 + |xj|^2 - 2*(xi . xj); the Gram term per 16x16 tile is
// one V_WMMA_F32_16X16X4_F32 (D=3 zero-padded to K=4). Bandwidth-bound on adj
// (128 MB @ 23.3 TB/s ~ 5.5us), so adj is loaded nontemporal, coalesced in the
// WMMA C-matrix lane layout. Raw v_sqrt_f32 (no denorm fixup) keeps the VALU
// path under the HBM roofline.

typedef float v2f __attribute__((ext_vector_type(2)));
typedef float v8f __attribute__((ext_vector_type(8)));

#define NPTS 2048
#define BATCH 8
#define JTILES_PER_WAVE 8   // tasks = 8 * 128 i-tiles * 16 j-chunks = 16384 waves

__global__ __launch_bounds__(256)
void idla_main(const float* __restrict__ preds,
               const float* __restrict__ targets,
               const float* __restrict__ adj,
               float* __restrict__ ws)
{
    const int lane = threadIdx.x & 31;
    const int wave = threadIdx.x >> 5;
    const int half = lane >> 4;     // 0: K0/K1 slots, 1: K2/K3 slots
    const int col  = lane & 15;

    // task decode: per batch 128 i-tiles * 16 j-chunks = 2048 tasks
    const int task    = blockIdx.x * 8 + wave;
    const int b       = task >> 11;
    const int rem     = task & 2047;
    const int i_tile  = rem >> 4;
    const int j_chunk = rem & 15;
    const int i0      = i_tile << 4;

    const float* __restrict__ pB = preds   + (size_t)b * NPTS * 3;
    const float* __restrict__ tB = targets + (size_t)b * NPTS * 3;
    const float* __restrict__ aB = adj     + ((size_t)b << 22);   // b * N * N

    // ---- A fragments (rows i0..i0+15), loop-invariant ----
    // 32-bit A 16x4 layout: lanes 0-15 -> (K0,K1), lanes 16-31 -> (K2,K3=0)
    v2f aP, aT;
    {
        const int n = (i0 + col) * 3;
        const float p0 = pB[n], p1 = pB[n + 1], p2 = pB[n + 2];
        const float t0 = tB[n], t1 = tB[n + 1], t2 = tB[n + 2];
        aP.x = half ? p2 : p0;  aP.y = half ? 0.0f : p1;
        aT.x = half ? t2 : t0;  aT.y = half ? 0.0f : t1;
    }

    // ---- row squared norms for this lane's C slots (M = v + 8*half) ----
    float sqiP[8], sqiT[8];
#pragma unroll
    for (int v = 0; v < 8; ++v) {
        const int r = (i0 + v + (half << 3)) * 3;
        const float p0 = pB[r], p1 = pB[r + 1], p2 = pB[r + 2];
        const float t0 = tB[r], t1 = tB[r + 1], t2 = tB[r + 2];
        sqiP[v] = fmaf(p0, p0, fmaf(p1, p1, p2 * p2));
        sqiT[v] = fmaf(t0, t0, fmaf(t1, t1, t2 * t2));
    }

    float acc  = 0.0f;   // sum of ((dp - dt) * adj)^2
    float asum = 0.0f;   // sum of adj

    for (int jj = 0; jj < JTILES_PER_WAVE; ++jj) {
        const int j0 = ((j_chunk << 3) + jj) << 4;

        // ---- B fragments (cols j0..j0+15), same K-split layout as A ----
        const int n = (j0 + col) * 3;
        const float p0 = pB[n], p1 = pB[n + 1], p2 = pB[n + 2];
        const float t0 = tB[n], t1 = tB[n + 1], t2 = tB[n + 2];
        v2f bP, bT;
        bP.x = half ? p2 : p0;  bP.y = half ? 0.0f : p1;
        bT.x = half ? t2 : t0;  bT.y = half ? 0.0f : t1;
        const float sqjP = fmaf(p0, p0, fmaf(p1, p1, p2 * p2));
        const float sqjT = fmaf(t0, t0, fmaf(t1, t1, t2 * t2));

        // ---- Gram tiles: one f32 WMMA each ----
        v8f cz = {};
        v8f gP = __builtin_amdgcn_wmma_f32_16x16x4_f32(
            false, aP, false, bP, (short)0, cz, false, false);
        v8f gT = __builtin_amdgcn_wmma_f32_16x16x4_f32(
            false, aT, false, bT, (short)0, cz, false, false);

        // adj in C-matrix layout: element (M = v + 8*half, N = col)
        const float* __restrict__ ap =
            aB + (size_t)(i0 + (half << 3)) * NPTS + (j0 + col);

#pragma unroll
        for (int v = 0; v < 8; ++v) {
            const float a   = __builtin_nontemporal_load(ap + (size_t)v * NPTS);
            float d2p = fmaf(-2.0f, gP[v], sqiP[v] + sqjP);
            float d2t = fmaf(-2.0f, gT[v], sqiT[v] + sqjT);
            d2p = fmaxf(d2p, 0.0f);
            d2t = fmaxf(d2t, 0.0f);
            // raw v_sqrt_f32: no denorm-range fixup sequence
            const float diff = __builtin_amdgcn_sqrtf(d2p)
                             - __builtin_amdgcn_sqrtf(d2t);
            const float m    = diff * a;
            acc  = fmaf(m, m, acc);
            asum += a;
        }
    }

    // ---- block reduction (256 threads) then 2 global atomics ----
    __shared__ float s_acc[256];
    __shared__ float s_sum[256];
    s_acc[threadIdx.x] = acc;
    s_sum[threadIdx.x] = asum;
    __syncthreads();
#pragma unroll
    for (int s = 128; s > 0; s >>= 1) {
        if ((int)threadIdx.x < s) {
            s_acc[threadIdx.x] += s_acc[threadIdx.x + s];
            s_sum[threadIdx.x] += s_sum[threadIdx.x + s];
        }
        __syncthreads();
    }
    if (threadIdx.x == 0) {
        atomicAdd(&ws[0], s_acc[0]);
        atomicAdd(&ws[1], s_sum[0]);
    }
}

__global__ void idla_finalize(const float* __restrict__ ws,
                              float* __restrict__ out)
{
    out[0] = ws[0] / ws[1];
}

extern "C" void kernel_launch(void* const* d_in, const int* in_sizes, int n_in,
                              void* d_out, int out_size, void* d_ws, size_t ws_size,
                              hipStream_t stream)
{
    const float* preds   = (const float*)d_in[0];
    const float* targets = (const float*)d_in[1];
    const float* adj     = (const float*)d_in[2];
    float* out = (float*)d_out;
    float* ws  = (float*)d_ws;

    hipMemsetAsync(ws, 0, 2 * sizeof(float), stream);

    // tasks = B * (N/16 i-tiles) * (N/16/8 j-chunks) = 8 * 128 * 16 = 16384 waves
    const int tasks  = BATCH * (NPTS / 16) * ((NPTS / 16) / JTILES_PER_WAVE);
    const int blocks = tasks / 8;   // 8 waves per 256-thread block
    idla_main<<<blocks, 256, 0, stream>>>(preds, targets, adj, ws);
    idla_finalize<<<1, 1, 0, stream>>>(ws, out);
}